// Multi_Match_41704132444282
// MI455X (gfx1250) — compile-verified
//
#include <hip/hip_runtime.h>
#include <hip/hip_bf16.h>

#define B_   16
#define L_   512
#define D_   256
#define P_   20
#define PM_  80
#define BLr  (B_ * L_)
#define EPSC 1e-8f
#define NEGC -10000000.0f

typedef float v2f __attribute__((ext_vector_type(2)));
typedef float v8f __attribute__((ext_vector_type(8)));

static __device__ __forceinline__ v8f wmma4(v2f a, v2f b, v8f c) {
  // D = A(16x4, f32) * B(4x16, f32) + C(16x16, f32)
  return __builtin_amdgcn_wmma_f32_16x16x4_f32(false, a, false, b, (short)0, c,
                                               false, false);
}

// ---------------------------------------------------------------------------
// Kernel 1: per-row plain norm + weighted norms for w1..w4 (both sides).
// norms[side*BL + row]; nw[(wi*2+side)*BL*P + row*P + p]
// ---------------------------------------------------------------------------
__global__ void k_row_norms(const float* __restrict__ a,
                            const float* __restrict__ bmat,
                            const float* __restrict__ w1,
                            const float* __restrict__ w2,
                            const float* __restrict__ w3,
                            const float* __restrict__ w4,
                            float* __restrict__ norms,
                            float* __restrict__ nw) {
  int row  = blockIdx.x % BLr;
  int side = blockIdx.x / BLr;
  const float* x = (side == 0 ? a : bmat) + (size_t)row * D_;
  __shared__ float x2[D_];
  int t = threadIdx.x;                     // 256 threads
  float xv = x[t];
  x2[t] = xv * xv;
  __syncthreads();
  if (t < 4 * P_) {
    int wi = t / P_, p = t % P_;
    const float* w = (wi == 0 ? w1 : wi == 1 ? w2 : wi == 2 ? w3 : w4) + p * D_;
    float s = 0.f;
    for (int d = 0; d < D_; ++d) {
      float wv = w[d];
      s += x2[d] * wv * wv;
    }
    nw[((size_t)(wi * 2 + side)) * ((size_t)BLr * P_) + (size_t)row * P_ + p] =
        sqrtf(s);
  } else if (t == 4 * P_) {
    float s = 0.f;
    for (int d = 0; d < D_; ++d) s += x2[d];
    norms[(size_t)side * BLr + row] = sqrtf(s);
  }
}

// ---------------------------------------------------------------------------
// Kernel 2: raw[b,i,j] = sum_d a[b,i,d] * b[b,j,d]   (WMMA f32 16x16x4)
// grid (L/64, L/32, B), block 256 (8 waves -> 4x2 grid of 16x16 tiles)
// ---------------------------------------------------------------------------
__global__ void k_raw_gemm(const float* __restrict__ a,
                           const float* __restrict__ bmat,
                           float* __restrict__ raw) {
  int batch = blockIdx.z;
  int i0 = blockIdx.x * 64;
  int j0 = blockIdx.y * 32;
  const float* A  = a    + (size_t)batch * L_ * D_;
  const float* Bm = bmat + (size_t)batch * L_ * D_;
  __shared__ float At[64][36];
  __shared__ float Bt[32][36];
  int t = threadIdx.x;
  int lane = t & 31, wid = t >> 5;
  int wr = wid & 3, wc = wid >> 2;
  int half = lane >> 4, l16 = lane & 15;
  v8f acc = {};
  for (int kc = 0; kc < D_; kc += 32) {
#pragma unroll
    for (int e = 0; e < 8; ++e) {
      int flat = t + e * 256;
      int r = flat >> 5, c = flat & 31;
      At[r][c] = A[(size_t)(i0 + r) * D_ + kc + c];
    }
#pragma unroll
    for (int e = 0; e < 4; ++e) {
      int flat = t + e * 256;
      int n = flat >> 5, k = flat & 31;
      Bt[k][n] = Bm[(size_t)(j0 + n) * D_ + kc + k];  // B[k][n] = b[n][k]
    }
    __syncthreads();
#pragma unroll
    for (int k4 = 0; k4 < 32; k4 += 4) {
      v2f af, bf;
      af.x = At[wr * 16 + l16][k4 + 2 * half];
      af.y = At[wr * 16 + l16][k4 + 2 * half + 1];
      bf.x = Bt[k4 + 2 * half][wc * 16 + l16];
      bf.y = Bt[k4 + 2 * half + 1][wc * 16 + l16];
      acc = wmma4(af, bf, acc);
    }
    __syncthreads();
  }
  float* out = raw + (size_t)batch * L_ * L_;
#pragma unroll
  for (int v = 0; v < 8; ++v) {
    int rr = wr * 16 + v + 8 * half;
    int cc = j0 + wc * 16 + l16;
    out[(size_t)(i0 + rr) * L_ + cc] = acc[v];
  }
}

// ---------------------------------------------------------------------------
// Kernel 3: per-m stats over k of raw: masked-softmax max & sum(exp),
// alpha max & sum.  TR=0: m=i, k=j (row stats); TR=1: m=j, k=i (col stats)
// One wave32 per m.
// ---------------------------------------------------------------------------
template <int TR>
__global__ void k_stats(const float* __restrict__ raw,
                        const int* __restrict__ mask_m,
                        const int* __restrict__ mask_k,
                        const float* __restrict__ nrm_m,
                        const float* __restrict__ nrm_k,
                        const float* __restrict__ tptr,
                        float* __restrict__ smax, float* __restrict__ ssum,
                        float* __restrict__ amaxo, float* __restrict__ asumo) {
  int wid = threadIdx.x >> 5, lane = threadIdx.x & 31;
  int gm = blockIdx.x * 8 + wid;           // 0..BL-1
  int batch = gm / L_, m = gm % L_;
  const float temp = *tptr;
  const float* R = raw + (size_t)batch * L_ * L_;
  float nm = nrm_m[gm];
  bool mk_m = mask_m[gm] != 0;
  float vals[16];
  float mmax = -3.0e38f, amx = -3.0e38f, asum_l = 0.f;
#pragma unroll
  for (int it = 0; it < 16; ++it) {
    int k = lane + it * 32;
    float r = (TR == 0) ? R[(size_t)m * L_ + k] : R[(size_t)k * L_ + m];
    bool msk = mk_m && (mask_k[batch * L_ + k] != 0);
    float attnv = msk ? r * temp : NEGC;
    float nk = nrm_k[batch * L_ + k];
    float al = r / (nm * nk + EPSC);
    vals[it] = attnv;
    mmax = fmaxf(mmax, attnv);
    amx = fmaxf(amx, al);
    asum_l += al;
  }
#pragma unroll
  for (int off = 16; off; off >>= 1) {
    mmax = fmaxf(mmax, __shfl_xor(mmax, off, 32));
    amx  = fmaxf(amx,  __shfl_xor(amx,  off, 32));
    asum_l += __shfl_xor(asum_l, off, 32);
  }
  float es = 0.f;
#pragma unroll
  for (int it = 0; it < 16; ++it) es += __expf(vals[it] - mmax);
#pragma unroll
  for (int off = 16; off; off >>= 1) es += __shfl_xor(es, off, 32);
  if (lane == 0) {
    smax[gm] = mmax;
    ssum[gm] = es;
    amaxo[gm] = amx;
    asumo[gm] = asum_l;
  }
}

// ---------------------------------------------------------------------------
// Kernel 4: out[b,m,:] = sum_k W(m,k) * Y[b,k,:]  (M=L, N=D, K=L)
// MODE 0: W = softmax weight (from raw + stats); MODE 1: W = alpha, then /S[m]
// TR selects raw[m,k] vs raw[k,m].
// ---------------------------------------------------------------------------
template <int MODE, int TR>
__global__ void k_wgemm(const float* __restrict__ raw,
                        const float* __restrict__ Y, float* __restrict__ out,
                        const int* __restrict__ mask_m,
                        const int* __restrict__ mask_k,
                        const float* __restrict__ smax,
                        const float* __restrict__ ssum,
                        const float* __restrict__ nrm_m,
                        const float* __restrict__ nrm_k,
                        const float* __restrict__ S,
                        const float* __restrict__ tptr) {
  int batch = blockIdx.z;
  int m0 = blockIdx.x * 64, n0 = blockIdx.y * 32;
  const float temp = (MODE == 0) ? *tptr : 0.f;
  const float* R  = raw + (size_t)batch * L_ * L_;
  const float* Yb = Y   + (size_t)batch * L_ * D_;
  __shared__ float At[64][36];
  __shared__ float Bt[32][36];
  int t = threadIdx.x;
  int lane = t & 31, wid = t >> 5;
  int wr = wid & 3, wc = wid >> 2;
  int half = lane >> 4, l16 = lane & 15;
  v8f acc = {};
  for (int kc = 0; kc < L_; kc += 32) {
#pragma unroll
    for (int e = 0; e < 8; ++e) {
      int flat = t + e * 256;
      int r, c;
      float rv;
      if (TR == 0) {
        r = flat >> 5; c = flat & 31;
        rv = R[(size_t)(m0 + r) * L_ + kc + c];
      } else {
        r = flat & 63; c = flat >> 6;
        rv = R[(size_t)(kc + c) * L_ + m0 + r];
      }
      int gm = batch * L_ + m0 + r;
      int gk = batch * L_ + kc + c;
      float w;
      if (MODE == 0) {
        bool msk = (mask_m[gm] != 0) && (mask_k[gk] != 0);
        float attnv = msk ? rv * temp : NEGC;
        w = __expf(attnv - smax[gm]) / ssum[gm];
      } else {
        w = rv / (nrm_m[gm] * nrm_k[gk] + EPSC);
      }
      At[r][c] = w;
    }
#pragma unroll
    for (int e = 0; e < 4; ++e) {
      int flat = t + e * 256;
      int k = flat >> 5, n = flat & 31;
      Bt[k][n] = Yb[(size_t)(kc + k) * D_ + n0 + n];
    }
    __syncthreads();
#pragma unroll
    for (int k4 = 0; k4 < 32; k4 += 4) {
      v2f af, bf;
      af.x = At[wr * 16 + l16][k4 + 2 * half];
      af.y = At[wr * 16 + l16][k4 + 2 * half + 1];
      bf.x = Bt[k4 + 2 * half][wc * 16 + l16];
      bf.y = Bt[k4 + 2 * half + 1][wc * 16 + l16];
      acc = wmma4(af, bf, acc);
    }
    __syncthreads();
  }
#pragma unroll
  for (int v = 0; v < 8; ++v) {
    int row = m0 + wr * 16 + v + 8 * half;
    int col = n0 + wc * 16 + l16;
    float res = acc[v];
    if (MODE == 1) res /= S[batch * L_ + row];
    out[((size_t)batch * L_ + row) * D_ + col] = res;
  }
}

// ---------------------------------------------------------------------------
// Kernel 5: mm[b,i,p] = max_j cos(b,i,j,p) with w2-weighted cosine.
// grid (L/64, P, B). Writes directly into m-output columns [20..39].
// ---------------------------------------------------------------------------
__global__ void k_mm(const float* __restrict__ X, const float* __restrict__ Y,
                     const float* __restrict__ nX2,
                     const float* __restrict__ nY2,
                     const float* __restrict__ w2, float* __restrict__ mout) {
  int batch = blockIdx.z, p = blockIdx.y;
  int i0 = blockIdx.x * 64;
  const float* Xb = X + (size_t)batch * L_ * D_;
  const float* Yb = Y + (size_t)batch * L_ * D_;
  const float* wp = w2 + p * D_;
  __shared__ float At[64][36];
  __shared__ float Bt[32][36];
  __shared__ float rmax_s[2][64];
  int t = threadIdx.x;
  int lane = t & 31, wid = t >> 5;
  int wr = wid & 3, wc = wid >> 2;
  int half = lane >> 4, l16 = lane & 15;
  float rowmax[8];
#pragma unroll
  for (int v = 0; v < 8; ++v) rowmax[v] = -3.0e38f;

  for (int j0 = 0; j0 < L_; j0 += 32) {
    v8f acc = {};
    for (int kc = 0; kc < D_; kc += 32) {
#pragma unroll
      for (int e = 0; e < 8; ++e) {
        int flat = t + e * 256;
        int r = flat >> 5, c = flat & 31;
        float wv = wp[kc + c];
        At[r][c] = Xb[(size_t)(i0 + r) * D_ + kc + c] * wv * wv;
      }
#pragma unroll
      for (int e = 0; e < 4; ++e) {
        int flat = t + e * 256;
        int n = flat >> 5, k = flat & 31;
        Bt[k][n] = Yb[(size_t)(j0 + n) * D_ + kc + k];
      }
      __syncthreads();
#pragma unroll
      for (int k4 = 0; k4 < 32; k4 += 4) {
        v2f af, bf;
        af.x = At[wr * 16 + l16][k4 + 2 * half];
        af.y = At[wr * 16 + l16][k4 + 2 * half + 1];
        bf.x = Bt[k4 + 2 * half][wc * 16 + l16];
        bf.y = Bt[k4 + 2 * half + 1][wc * 16 + l16];
        acc = wmma4(af, bf, acc);
      }
      __syncthreads();
    }
#pragma unroll
    for (int v = 0; v < 8; ++v) {
      int gi = i0 + wr * 16 + v + 8 * half;
      int gj = j0 + wc * 16 + l16;
      float denom = nX2[((size_t)batch * L_ + gi) * P_ + p] *
                        nY2[((size_t)batch * L_ + gj) * P_ + p] +
                    EPSC;
      rowmax[v] = fmaxf(rowmax[v], acc[v] / denom);
    }
  }
  // reduce across the 16 lanes that share each row (within a half-group)
#pragma unroll
  for (int off = 8; off; off >>= 1) {
#pragma unroll
    for (int v = 0; v < 8; ++v)
      rowmax[v] = fmaxf(rowmax[v], __shfl_xor(rowmax[v], off, 32));
  }
  if (l16 == 0) {
#pragma unroll
    for (int v = 0; v < 8; ++v)
      rmax_s[wc][wr * 16 + 8 * half + v] = rowmax[v];
  }
  __syncthreads();
  if (t < 64) {
    float mval = fmaxf(rmax_s[0][t], rmax_s[1][t]);
    mout[((size_t)batch * L_ + i0 + t) * PM_ + 20 + p] = mval;
  }
}

// ---------------------------------------------------------------------------
// Kernel 6: assemble fm / am / mam columns of m_a / m_b.
// ---------------------------------------------------------------------------
__global__ void k_final(const float* __restrict__ a,
                        const float* __restrict__ bmat,
                        const float* __restrict__ s2m,
                        const float* __restrict__ s1m,
                        const float* __restrict__ w1,
                        const float* __restrict__ w3,
                        const float* __restrict__ w4,
                        const float* __restrict__ nw,
                        const float* __restrict__ amaxR,
                        const float* __restrict__ amaxC,
                        float* __restrict__ m_a, float* __restrict__ m_b) {
  int row  = blockIdx.x % BLr;   // b*L + l
  int side = blockIdx.x / BLr;
  int batch = row / L_;
  const float* x = (side == 0 ? a : bmat) + (size_t)row * D_;
  const float* s = (side == 0 ? s2m : s1m) + (size_t)row * D_;
  const float* lastOther =
      (side == 0 ? bmat : a) + ((size_t)batch * L_ + (L_ - 1)) * D_;
  float amax = (side == 0 ? amaxR : amaxC)[row];
  float* mo = (side == 0 ? m_a : m_b) + (size_t)row * PM_;
  __shared__ float xs[D_], ssh[D_], ls[D_];
  int t = threadIdx.x;
  xs[t] = x[t];
  ssh[t] = s[t];
  ls[t] = lastOther[t];
  __syncthreads();
  if (t < P_) {
    int p = t;
    const float* w1p = w1 + p * D_;
    const float* w3p = w3 + p * D_;
    const float* w4p = w4 + p * D_;
    float fmn = 0.f, amn = 0.f, ns2 = 0.f, mamn = 0.f, wn4 = 0.f;
    for (int d = 0; d < D_; ++d) {
      float w1s = w1p[d] * w1p[d];
      float w3s = w3p[d] * w3p[d];
      float w4s = w4p[d] * w4p[d];
      float xv = xs[d], sv = ssh[d], lv = ls[d];
      fmn  += xv * w1s * lv;
      amn  += xv * w3s * sv;
      ns2  += sv * sv * w3s;
      mamn += xv * w4s;
      wn4  += w4s;
    }
    ns2 = sqrtf(ns2);
    wn4 = sqrtf(wn4);
    size_t BLP = (size_t)BLr * P_;
    float n1 = nw[(0 * 2 + side) * BLP + (size_t)row * P_ + p];
    float n3 = nw[(2 * 2 + side) * BLP + (size_t)row * P_ + p];
    float n4 = nw[(3 * 2 + side) * BLP + (size_t)row * P_ + p];
    float nl1 = nw[(0 * 2 + (1 - side)) * BLP +
                   ((size_t)batch * L_ + (L_ - 1)) * (size_t)P_ + p];
    mo[p]      = fmn / fmaxf(n1 * nl1, EPSC);
    mo[40 + p] = amn / fmaxf(n3 * ns2, EPSC);
    mo[60 + p] = amax * mamn / fmaxf(n4 * fabsf(amax) * wn4, EPSC);
  }
}

// ---------------------------------------------------------------------------
extern "C" void kernel_launch(void* const* d_in, const int* in_sizes, int n_in,
                              void* d_out, int out_size, void* d_ws,
                              size_t ws_size, hipStream_t stream) {
  (void)in_sizes; (void)n_in; (void)out_size; (void)ws_size;
  const float* a      = (const float*)d_in[0];
  const float* bmat   = (const float*)d_in[1];
  const int*   mask_a = (const int*)d_in[2];
  const int*   mask_b = (const int*)d_in[3];
  const float* w1     = (const float*)d_in[4];
  const float* w2     = (const float*)d_in[5];
  const float* w3     = (const float*)d_in[6];
  const float* w4     = (const float*)d_in[7];
  const float* tptr   = (const float*)d_in[8];

  float* ia  = (float*)d_out;
  float* ib  = ia + (size_t)BLr * D_;
  float* m_a = ib + (size_t)BLr * D_;
  float* m_b = m_a + (size_t)BLr * PM_;

  float* wsf = (float*)d_ws;
  size_t off = 0;
  float* rawb  = wsf + off; off += (size_t)B_ * L_ * L_;
  float* norms = wsf + off; off += 2 * (size_t)BLr;     // A then B
  float* nw    = wsf + off; off += 8 * (size_t)BLr * P_;
  float* smaxR = wsf + off; off += BLr;
  float* ssumR = wsf + off; off += BLr;
  float* smaxC = wsf + off; off += BLr;
  float* ssumC = wsf + off; off += BLr;
  float* amaxR = wsf + off; off += BLr;
  float* asumR = wsf + off; off += BLr;
  float* amaxC = wsf + off; off += BLr;
  float* asumC = wsf + off; off += BLr;
  float* s2m   = wsf + off; off += (size_t)BLr * D_;
  float* s1m   = wsf + off; off += (size_t)BLr * D_;

  const float* normA = norms;
  const float* normB = norms + BLr;
  const size_t BLP = (size_t)BLr * P_;
  const float* nw2a = nw + (1 * 2 + 0) * BLP;
  const float* nw2b = nw + (1 * 2 + 1) * BLP;

  // 1. norms
  k_row_norms<<<2 * BLr, 256, 0, stream>>>(a, bmat, w1, w2, w3, w4, norms, nw);
  // 2. raw = a @ b^T
  k_raw_gemm<<<dim3(L_ / 64, L_ / 32, B_), 256, 0, stream>>>(a, bmat, rawb);
  // 3. stats
  k_stats<0><<<BLr / 8, 256, 0, stream>>>(rawb, mask_a, mask_b, normA, normB,
                                          tptr, smaxR, ssumR, amaxR, asumR);
  k_stats<1><<<BLr / 8, 256, 0, stream>>>(rawb, mask_b, mask_a, normB, normA,
                                          tptr, smaxC, ssumC, amaxC, asumC);
  // 4. attention outputs
  dim3 gw(L_ / 64, D_ / 32, B_);
  k_wgemm<0, 0><<<gw, 256, 0, stream>>>(rawb, bmat, ia, mask_a, mask_b, smaxR,
                                        ssumR, nullptr, nullptr, nullptr, tptr);
  k_wgemm<0, 1><<<gw, 256, 0, stream>>>(rawb, a, ib, mask_b, mask_a, smaxC,
                                        ssumC, nullptr, nullptr, nullptr, tptr);
  // 5. alpha-weighted means
  k_wgemm<1, 0><<<gw, 256, 0, stream>>>(rawb, bmat, s2m, nullptr, nullptr,
                                        nullptr, nullptr, normA, normB, asumR,
                                        tptr);
  k_wgemm<1, 1><<<gw, 256, 0, stream>>>(rawb, a, s1m, nullptr, nullptr, nullptr,
                                        nullptr, normB, normA, asumC, tptr);
  // 6. mm (cos max) columns
  k_mm<<<dim3(L_ / 64, P_, B_), 256, 0, stream>>>(a, bmat, nw2a, nw2b, w2, m_a);
  k_mm<<<dim3(L_ / 64, P_, B_), 256, 0, stream>>>(bmat, a, nw2b, nw2a, w2, m_b);
  // 7. fm / am / mam columns
  k_final<<<2 * BLr, 256, 0, stream>>>(a, bmat, s2m, s1m, w1, w3, w4, nw, amaxR,
                                       amaxC, m_a, m_b);
}